// BPModel_60086592471432
// MI455X (gfx1250) — compile-verified
//

#include <hip/hip_runtime.h>
#include <hip/hip_bf16.h>

// -------- CDNA5 wave32 / WMMA types --------
typedef __attribute__((ext_vector_type(2))) float v2f;
typedef __attribute__((ext_vector_type(8))) float v8f;

static constexpr int Cn = 8;          // num classes (channels)
static constexpr int NUM_ITERS = 16;
static constexpr float THRESHOLD = 1e-8f;

struct Hdr {
    unsigned int diff_bits;   // max |delta beliefs| as float bits (non-negative -> uint order == float order)
    int converged;
};

// ds_swizzle helpers (group-of-32 mode, offset[15]=0):
//   src_lane = ((lane & and_mask) | or_mask) ^ xor_mask
//   offset = (xor<<10) | (or<<5) | and
__device__ __forceinline__ float swz_bcast_sum(float x) {
    // src_lane = (lane & 0x10) | 8  -> broadcast lane {8,24} to each 16-lane half
    return __int_as_float(__builtin_amdgcn_ds_swizzle(__float_as_int(x), 0x0110));
}
#define SWZ_XOR(x, mask) \
    __int_as_float(__builtin_amdgcn_ds_swizzle(__float_as_int(x), (((mask) << 10) | 0x1F)))

// ---------------------------------------------------------------------------
// init: messages = 1/C
__global__ void k_init_msg(float* __restrict__ msg, long n) {
    long i = (long)blockIdx.x * blockDim.x + threadIdx.x;
    long stride = (long)gridDim.x * blockDim.x;
    for (; i < n; i += stride) msg[i] = 1.0f / Cn;
}

// init: log_priors, beliefs = priors, header, extended potential
// Pext[8][16]: cols 0..7 = P, col 8 = row-sum of P (normalization denominator
// falls out of the WMMA), cols 9..15 = 0.
__global__ void k_init_node(const float* __restrict__ priors,
                            const float* __restrict__ pot,   // 8x8
                            float* __restrict__ Pext,        // 8x16
                            float* __restrict__ logp,
                            float* __restrict__ beliefs,
                            Hdr* hdr, int nc) {
    int i = blockIdx.x * blockDim.x + threadIdx.x;
    if (i == 0) { hdr->diff_bits = 0u; hdr->converged = 0; }
    if (i < 128) {
        int k = i >> 4, c = i & 15;
        float v = 0.f;
        if (c < Cn) {
            v = pot[k * Cn + c];
        } else if (c == Cn) {
            for (int j = 0; j < Cn; ++j) v += pot[k * Cn + j];
        }
        Pext[i] = v;
    }
    if (i < nc) {
        float p = priors[i];
        logp[i] = __logf(p);
        beliefs[i] = p;
    }
}

// ---------------------------------------------------------------------------
// Edge pass: one wave handles a tile of 16 edges via two chained
// V_WMMA_F32_16X16X4_F32 ops:  raw(16x9) = X(16x8) @ Pext(8x16), K split 4+4.
//   A VGPR j, lane L : X[L%16][2*(L/16)+j]
//   B VGPR j, lane L : Pext[2*(L/16)+j][L%16]
//   D VGPR v, lane L : raw[v + 8*(L/16)][L%16]   (col 8 = row sum)
__global__ void k_edges(const float* __restrict__ beliefs,
                        const float* __restrict__ msg_old,
                        float*       __restrict__ msg_new,
                        const float* __restrict__ Pext,   // 8x16 (see k_init_node)
                        const int*   __restrict__ src,
                        const int*   __restrict__ rev,
                        const Hdr*   __restrict__ hdr,
                        int ntiles) {
    if (hdr->converged) return;                       // wave-uniform exit
    int gtid = blockIdx.x * blockDim.x + threadIdx.x;
    int wave = gtid >> 5;
    if (wave >= ntiles) return;                       // wave-uniform (full blocks)
    int lane = threadIdx.x & 31;
    int m = lane & 15;            // edge row within tile / B column
    int p = lane >> 4;            // half-wave selector
    int base = wave << 4;
    int e = base + m;
    int s = src[e];
    int r = rev[e];

    const float* bp = beliefs + (long)s * Cn;
    const float* mp = msg_old + (long)r * Cn;
    int c0 = 2 * p;               // K base for this half-wave

    // A operands: x = beliefs[src]/msg_old[rev], channels (c0,c0+1) and (c0+4,c0+5)
    v2f a1, a2;
    a1.x = bp[c0]     * __builtin_amdgcn_rcpf(mp[c0]);
    a1.y = bp[c0 + 1] * __builtin_amdgcn_rcpf(mp[c0 + 1]);
    a2.x = bp[c0 + 4] * __builtin_amdgcn_rcpf(mp[c0 + 4]);
    a2.y = bp[c0 + 5] * __builtin_amdgcn_rcpf(mp[c0 + 5]);

    // B operands: Pext rows (K) x column (lane%16); unconditional, EXEC stays full.
    v2f b1, b2;
    b1.x = Pext[(c0)     * 16 + m];
    b1.y = Pext[(c0 + 1) * 16 + m];
    b2.x = Pext[(c0 + 4) * 16 + m];
    b2.y = Pext[(c0 + 5) * 16 + m];

    v8f acc = {};
    acc = __builtin_amdgcn_wmma_f32_16x16x4_f32(false, a1, false, b1, (short)0, acc, false, false);
    acc = __builtin_amdgcn_wmma_f32_16x16x4_f32(false, a2, false, b2, (short)0, acc, false, false);

    // Column 8 of D holds the row sum: one swizzle broadcast per D VGPR.
    float outv[8];
#pragma unroll
    for (int v = 0; v < 8; ++v) {
        float val = acc[v];
        float rs = swz_bcast_sum(val);
        outv[v] = val * __builtin_amdgcn_rcpf(rs);
    }
    if (m < Cn) {   // lanes holding valid channels store 8 edge rows each
        float* dst = msg_new + (long)(base + 8 * p) * Cn + m;
#pragma unroll
        for (int v = 0; v < 8; ++v) dst[(long)v * Cn] = outv[v];
    }
}

// Scalar fallback for a trailing partial tile (E % 16 != 0; unused for this setup).
__global__ void k_edges_tail(const float* __restrict__ beliefs,
                             const float* __restrict__ msg_old,
                             float*       __restrict__ msg_new,
                             const float* __restrict__ Pext,  // 8x16
                             const int*   __restrict__ src,
                             const int*   __restrict__ rev,
                             const Hdr*   __restrict__ hdr,
                             int estart, int E) {
    if (hdr->converged) return;
    int e = estart + blockIdx.x * blockDim.x + threadIdx.x;
    if (e >= E) return;
    int s = src[e], r = rev[e];
    float x[Cn], raw[Cn];
#pragma unroll
    for (int c = 0; c < Cn; ++c)
        x[c] = beliefs[(long)s * Cn + c] * __builtin_amdgcn_rcpf(msg_old[(long)r * Cn + c]);
    float ssum = 0.f;
#pragma unroll
    for (int j = 0; j < Cn; ++j) {
        float acc = 0.f;
#pragma unroll
        for (int i = 0; i < Cn; ++i) acc += x[i] * Pext[i * 16 + j];
        raw[j] = acc; ssum += acc;
    }
    float inv = __builtin_amdgcn_rcpf(ssum);
#pragma unroll
    for (int j = 0; j < Cn; ++j) msg_new[(long)e * Cn + j] = raw[j] * inv;
}

// ---------------------------------------------------------------------------
// Node pass: incoming edges of node n are rev[n*deg .. n*deg+deg-1].
// logb = log_priors + sum log(msg); beliefs = softmax(logb); track max diff.
__global__ void k_beliefs(const float* __restrict__ msg,
                          const float* __restrict__ logp,
                          const int*   __restrict__ rev,
                          float*       __restrict__ beliefs,
                          Hdr* hdr, int N, int deg) {
    if (hdr->converged) return;
    int n = blockIdx.x * blockDim.x + threadIdx.x;
    float localdiff = 0.f;
    if (n < N) {
        float acc[Cn];
#pragma unroll
        for (int c = 0; c < Cn; ++c) acc[c] = logp[(long)n * Cn + c];
        const int* rv = rev + (long)n * deg;
        for (int j = 0; j < deg; ++j) {
            long e = rv[j];
            const float4* mp = (const float4*)(msg + e * Cn);
            float4 lo = mp[0], hi = mp[1];
            acc[0] += __logf(lo.x); acc[1] += __logf(lo.y);
            acc[2] += __logf(lo.z); acc[3] += __logf(lo.w);
            acc[4] += __logf(hi.x); acc[5] += __logf(hi.y);
            acc[6] += __logf(hi.z); acc[7] += __logf(hi.w);
        }
        float mx = acc[0];
#pragma unroll
        for (int c = 1; c < Cn; ++c) mx = fmaxf(mx, acc[c]);
        float ex[Cn]; float s = 0.f;
#pragma unroll
        for (int c = 0; c < Cn; ++c) { ex[c] = __expf(acc[c] - mx); s += ex[c]; }
        float inv = __builtin_amdgcn_rcpf(s);
#pragma unroll
        for (int c = 0; c < Cn; ++c) {
            float nb = ex[c] * inv;
            float ob = beliefs[(long)n * Cn + c];
            localdiff = fmaxf(localdiff, fabsf(nb - ob));
            beliefs[(long)n * Cn + c] = nb;
        }
    }
    // wave32 butterfly max-reduce via ds_swizzle, then one atomic per wave
    localdiff = fmaxf(localdiff, SWZ_XOR(localdiff, 1));
    localdiff = fmaxf(localdiff, SWZ_XOR(localdiff, 2));
    localdiff = fmaxf(localdiff, SWZ_XOR(localdiff, 4));
    localdiff = fmaxf(localdiff, SWZ_XOR(localdiff, 8));
    localdiff = fmaxf(localdiff, SWZ_XOR(localdiff, 16));
    if ((threadIdx.x & 31) == 0)
        atomicMax(&hdr->diff_bits, __float_as_uint(localdiff));
}

__global__ void k_check(Hdr* hdr, float thr) {
    if (!hdr->converged && __uint_as_float(hdr->diff_bits) < thr) hdr->converged = 1;
    hdr->diff_bits = 0u;
}

// ---------------------------------------------------------------------------
extern "C" void kernel_launch(void* const* d_in, const int* in_sizes, int n_in,
                              void* d_out, int out_size, void* d_ws, size_t ws_size,
                              hipStream_t stream) {
    const float* priors = (const float*)d_in[0];   // [N, 8]
    const float* pot    = (const float*)d_in[1];   // [8, 8]
    const int*   src    = (const int*)d_in[2];     // [E]
    // d_in[3] = dst_nodes (unused: rev gives incoming-edge gather)
    const int*   rev    = (const int*)d_in[4];     // [E]

    const int NC = in_sizes[0];
    const int N  = NC / Cn;
    const int E  = in_sizes[2];
    const int deg = E / N;                         // 32
    const long EC = (long)E * Cn;

    char* ws = (char*)d_ws;
    Hdr*   hdr  = (Hdr*)ws;                        // 256 B
    float* Pext = (float*)(ws + 256);              // 8x16 floats = 512 B
    float* msgA = (float*)(ws + 768);
    float* msgB = (float*)(ws + 768 + EC * sizeof(float));
    float* logp = (float*)(ws + 768 + 2 * EC * sizeof(float));
    float* beliefs = (float*)d_out;

    // ---- init state (every call: ws/d_out are not re-poisoned between replays)
    k_init_msg<<<4096, 256, 0, stream>>>(msgA, EC);
    k_init_node<<<(NC + 255) / 256, 256, 0, stream>>>(priors, pot, Pext, logp, beliefs, hdr, NC);

    const int ntiles = E / 16;
    const int tail   = E - ntiles * 16;
    const int edgeBlocks = (ntiles + 7) / 8;       // 8 waves (of 32) per 256-thread block
    const int nodeBlocks = (N + 255) / 256;

    float* mold = msgA;
    float* mnew = msgB;
    for (int it = 0; it < NUM_ITERS; ++it) {
        k_edges<<<edgeBlocks, 256, 0, stream>>>(beliefs, mold, mnew, Pext, src, rev, hdr, ntiles);
        if (tail)
            k_edges_tail<<<(tail + 255) / 256, 256, 0, stream>>>(beliefs, mold, mnew, Pext, src, rev,
                                                                 hdr, ntiles * 16, E);
        k_beliefs<<<nodeBlocks, 256, 0, stream>>>(mnew, logp, rev, beliefs, hdr, N, deg);
        k_check<<<1, 1, 0, stream>>>(hdr, THRESHOLD);
        float* t = mold; mold = mnew; mnew = t;
    }
}
